// OmegaRestraint_81612968558776
// MI455X (gfx1250) — compile-verified
//
#include <hip/hip_runtime.h>
#include <math.h>

#define LDIM   512
#define NBINS  24
#define NCUT   (NBINS + 1)
#define BLK    256

#define AS1 __attribute__((address_space(1)))
#define AS3 __attribute__((address_space(3)))

typedef int v4i __attribute__((ext_vector_type(4)));

#if defined(__AMDGCN__) && __has_builtin(__builtin_amdgcn_global_load_async_to_lds_b128)
#define HAVE_ASYNC_LDS 1
#else
#define HAVE_ASYNC_LDS 0
#endif

// ---- small vector helpers -------------------------------------------------
struct V3 { float x, y, z; };
__device__ __forceinline__ V3 v3sub(V3 a, V3 b) { return {a.x - b.x, a.y - b.y, a.z - b.z}; }
__device__ __forceinline__ V3 v3cross(V3 a, V3 b) {
    return {a.y * b.z - a.z * b.y, a.z * b.x - a.x * b.z, a.x * b.y - a.y * b.x};
}
__device__ __forceinline__ float v3dot(V3 a, V3 b) { return a.x * b.x + a.y * b.y + a.z * b.z; }

// One 16-byte global -> LDS copy. On CDNA5 this lowers to
// global_load_async_to_lds_b128 (tracked by ASYNCcnt); fallback is a plain
// vector load + LDS store.
__device__ __forceinline__ void async_copy16(const float* __restrict__ gsrc, float* lds_dst) {
#if HAVE_ASYNC_LDS
    __builtin_amdgcn_global_load_async_to_lds_b128(
        (AS1 v4i*)gsrc, (AS3 v4i*)lds_dst, /*imm offset*/ 0, /*cpol*/ 0);
#else
    *reinterpret_cast<float4*>(lds_dst) = *reinterpret_cast<const float4*>(gsrc);
#endif
}

__device__ __forceinline__ void wait_async_lds() {
#if HAVE_ASYNC_LDS
#if __has_builtin(__builtin_amdgcn_s_wait_asynccnt)
    __builtin_amdgcn_s_wait_asynccnt(0);
#else
    asm volatile("s_wait_asynccnt 0" ::: "memory");
#endif
#endif
}

// ---- stage 1: dihedral + spline eval, per-block partial sums --------------
__global__ void __launch_bounds__(BLK)
omega_restraint_stage1(const float* __restrict__ CA,      // (B, L, 3)
                       const float* __restrict__ CB,      // (B, L, 3)
                       const float* __restrict__ coeff,   // (L, L, NBINS, 4)
                       const float* __restrict__ cutoffs, // (NCUT,)
                       const int*   __restrict__ x_idx,   // (npairs,)
                       const int*   __restrict__ y_idx,   // (npairs,)
                       float*       __restrict__ partials,
                       int npairs) {
    __shared__ float sCA[LDIM * 3];   // 6 KB: batch-b CA points
    __shared__ float sCB[LDIM * 3];   // 6 KB: batch-b CB points
    __shared__ float sCut[NCUT];
    __shared__ float sWave[BLK / 32];

    const int t = threadIdx.x;
    const int b = blockIdx.y;
    const float* gCA = CA + (size_t)b * LDIM * 3;
    const float* gCB = CB + (size_t)b * LDIM * 3;

    // 512*3 = 1536 floats = 384 float4 per array; 256 threads do up to 2 each.
#pragma unroll
    for (int k = 0; k < 2; ++k) {
        int q = t + k * BLK;          // float4 index, 0..383 valid
        if (q < (LDIM * 3) / 4) {
            async_copy16(gCA + q * 4, sCA + q * 4);
            async_copy16(gCB + q * 4, sCB + q * 4);
        }
    }
    if (t < NCUT) sCut[t] = cutoffs[t];
    wait_async_lds();
    __syncthreads();

    const int pair = blockIdx.x * BLK + t;
    float val = 0.0f;
    if (pair < npairs) {
        const int xi = x_idx[pair];
        const int yi = y_idx[pair];

        V3 p0 = {sCA[xi * 3 + 0], sCA[xi * 3 + 1], sCA[xi * 3 + 2]};
        V3 p1 = {sCB[xi * 3 + 0], sCB[xi * 3 + 1], sCB[xi * 3 + 2]};
        V3 p2 = {sCB[yi * 3 + 0], sCB[yi * 3 + 1], sCB[yi * 3 + 2]};
        V3 p3 = {sCA[yi * 3 + 0], sCA[yi * 3 + 1], sCA[yi * 3 + 2]};

        V3 b1 = v3sub(p1, p0);
        V3 b2 = v3sub(p2, p1);
        V3 b3 = v3sub(p3, p2);
        V3 n1 = v3cross(b1, b2);
        V3 n2 = v3cross(b2, b3);
        float inv = 1.0f / (sqrtf(v3dot(b2, b2)) + 1e-9f);
        V3 b2n = {b2.x * inv, b2.y * inv, b2.z * inv};
        V3 m1 = v3cross(n1, b2n);
        float om = atan2f(v3dot(m1, n2), v3dot(n1, n2));

        if (om < sCut[0]) om += 6.283185307179586f;

        // exact searchsorted(side='right') - 1, clipped to [0, NBINS-1]
        int cnt = 0;
#pragma unroll
        for (int k = 0; k < NCUT; ++k) cnt += (om >= sCut[k]) ? 1 : 0;
        int bin = cnt - 1;
        bin = bin < 0 ? 0 : (bin > NBINS - 1 ? NBINS - 1 : bin);

        const float dx = om - sCut[bin];
        const size_t off = ((((size_t)xi * LDIM + yi) * NBINS) + bin) * 4;  // 16B aligned
        const float4 c = *reinterpret_cast<const float4*>(coeff + off);
        val = fmaf(dx, fmaf(dx, fmaf(dx, c.w, c.z), c.y), c.x);
    }

    // deterministic reduction: wave32 shuffle tree, then fixed-order sum
#pragma unroll
    for (int offl = 16; offl > 0; offl >>= 1)
        val += __shfl_down(val, offl, 32);
    const int lane = t & 31, wave = t >> 5;
    if (lane == 0) sWave[wave] = val;
    __syncthreads();
    if (t == 0) {
        float s = 0.0f;
#pragma unroll
        for (int w = 0; w < BLK / 32; ++w) s += sWave[w];
        partials[(size_t)blockIdx.y * gridDim.x + blockIdx.x] = s;
    }
}

// ---- stage 2: deterministic final reduction -------------------------------
__global__ void __launch_bounds__(BLK)
omega_restraint_stage2(const float* __restrict__ partials, int n, float* __restrict__ out) {
    __shared__ float s[BLK];
    float v = 0.0f;
    for (int i = threadIdx.x; i < n; i += BLK) v += partials[i];  // fixed order
    s[threadIdx.x] = v;
    __syncthreads();
#pragma unroll
    for (int off = BLK / 2; off > 0; off >>= 1) {
        if (threadIdx.x < off) s[threadIdx.x] += s[threadIdx.x + off];
        __syncthreads();
    }
    if (threadIdx.x == 0) out[0] = s[0];
}

extern "C" void kernel_launch(void* const* d_in, const int* in_sizes, int n_in,
                              void* d_out, int out_size, void* d_ws, size_t ws_size,
                              hipStream_t stream) {
    const float* CA      = (const float*)d_in[0];  // (B, L, 3)
    const float* CB      = (const float*)d_in[1];  // (B, L, 3)
    const float* coeff   = (const float*)d_in[2];  // (L, L, NBINS, 4)
    const float* cutoffs = (const float*)d_in[3];  // (NCUT,)
    const int*   x_idx   = (const int*)d_in[4];
    const int*   y_idx   = (const int*)d_in[5];

    const int npairs = in_sizes[4];
    const int B      = in_sizes[0] / (LDIM * 3);
    const int gx     = (npairs + BLK - 1) / BLK;

    float* partials = (float*)d_ws;  // gx * B floats (~4 KB)

    dim3 grid(gx, B);
    omega_restraint_stage1<<<grid, BLK, 0, stream>>>(CA, CB, coeff, cutoffs,
                                                     x_idx, y_idx, partials, npairs);
    omega_restraint_stage2<<<1, BLK, 0, stream>>>(partials, gx * B, (float*)d_out);
}